// GMMModel_79937931313650
// MI455X (gfx1250) — compile-verified
//
#include <hip/hip_runtime.h>

typedef float v2f __attribute__((ext_vector_type(2)));
typedef float v8f __attribute__((ext_vector_type(8)));

static __device__ __forceinline__ v8f wmma4(v2f a, v2f b, v8f c) {
  return __builtin_amdgcn_wmma_f32_16x16x4_f32(false, a, false, b, (short)0, c,
                                               false, false);
}

// ---------------------------------------------------------------------------
// Generic fp32 WMMA GEMM: C = epi(A @ B); one wave computes a 32x16 C tile
// (two 16x16 accumulators sharing the B operand -> half the B traffic, and
// alternating acc0/acc1 gives independent back-to-back v_wmma issues).
// A: M x K row-major (or K x M if TRANSA: reads A[k*M+m]); B: K x N row-major.
// REQUIRES K % 4 == 0 (all call sites satisfy this).
// Edge tiles: row/col indices clamped for loads (garbage accumulates only in
// never-stored rows/cols) so the K-loop has no exec-mask manipulation.
// Batched over blockIdx.z with element strides sA/sB/sC.
// Epilogue: +bias (mode 1: per-col N, mode 2: per-row M), optional ReLU,
// then optional residual add from addsrc (stride sC), store to C.
// ---------------------------------------------------------------------------
template <int TRANSA>
__global__ __launch_bounds__(32) void k_wmma_gemm_f32(
    const float* __restrict__ A, const float* __restrict__ Bw,
    const float* __restrict__ bias, const float* __restrict__ addsrc,
    float* __restrict__ C, int M, int N, int K,
    long long sA, long long sB, long long sC,
    int bias_mode, int do_relu) {
  const int lane = threadIdx.x & 31;
  const long long z = blockIdx.z;
  const float* Ab = A + sA * z;
  const float* Bb = Bw + sB * z;
  const float* Addb = addsrc ? (addsrc + sC * z) : (const float*)0;
  float* Cb = C + sC * z;

  const int mt = blockIdx.x << 5;   // 32 rows per wave
  const int nt = blockIdx.y << 4;   // 16 cols per wave
  const int r15 = lane & 15;
  const int khalf = (lane >> 4) << 1;     // 0 or 2

  int am0 = mt + r15;      if (am0 > M - 1) am0 = M - 1;   // clamp (loads only)
  int am1 = mt + 16 + r15; if (am1 > M - 1) am1 = M - 1;
  int bn  = nt + r15;      if (bn  > N - 1) bn  = N - 1;

  const size_t Ms = (size_t)M, Ns = (size_t)N;
  const float* ap0;
  const float* ap1;
  if (TRANSA) {
    ap0 = Ab + (size_t)khalf * Ms + am0;
    ap1 = Ab + (size_t)khalf * Ms + am1;
  } else {
    ap0 = Ab + (size_t)am0 * K + khalf;
    ap1 = Ab + (size_t)am1 * K + khalf;
  }
  const float* bp = Bb + (size_t)khalf * Ns + bn;

  v8f acc0 = {0.f, 0.f, 0.f, 0.f, 0.f, 0.f, 0.f, 0.f};
  v8f acc1 = {0.f, 0.f, 0.f, 0.f, 0.f, 0.f, 0.f, 0.f};

  const int kiter = K >> 2;   // number of 16x16x4 steps
  int k = 0;
  for (; k + 4 <= kiter; k += 4) {
    v2f x0, x1, x2, x3, y0, y1, y2, y3, b0, b1, b2, b3;
    if (TRANSA) {
      x0.x = ap0[0];       x0.y = ap0[Ms];
      x1.x = ap0[4 * Ms];  x1.y = ap0[5 * Ms];
      x2.x = ap0[8 * Ms];  x2.y = ap0[9 * Ms];
      x3.x = ap0[12 * Ms]; x3.y = ap0[13 * Ms];
      ap0 += 16 * Ms;
      y0.x = ap1[0];       y0.y = ap1[Ms];
      y1.x = ap1[4 * Ms];  y1.y = ap1[5 * Ms];
      y2.x = ap1[8 * Ms];  y2.y = ap1[9 * Ms];
      y3.x = ap1[12 * Ms]; y3.y = ap1[13 * Ms];
      ap1 += 16 * Ms;
    } else {
      x0 = *(const v2f*)(ap0);
      x1 = *(const v2f*)(ap0 + 4);
      x2 = *(const v2f*)(ap0 + 8);
      x3 = *(const v2f*)(ap0 + 12);
      ap0 += 16;
      y0 = *(const v2f*)(ap1);
      y1 = *(const v2f*)(ap1 + 4);
      y2 = *(const v2f*)(ap1 + 8);
      y3 = *(const v2f*)(ap1 + 12);
      ap1 += 16;
    }
    b0.x = bp[0];       b0.y = bp[Ns];
    b1.x = bp[4 * Ns];  b1.y = bp[5 * Ns];
    b2.x = bp[8 * Ns];  b2.y = bp[9 * Ns];
    b3.x = bp[12 * Ns]; b3.y = bp[13 * Ns];
    bp += 16 * Ns;
    acc0 = wmma4(x0, b0, acc0);
    acc1 = wmma4(y0, b0, acc1);
    acc0 = wmma4(x1, b1, acc0);
    acc1 = wmma4(y1, b1, acc1);
    acc0 = wmma4(x2, b2, acc0);
    acc1 = wmma4(y2, b2, acc1);
    acc0 = wmma4(x3, b3, acc0);
    acc1 = wmma4(y3, b3, acc1);
  }
  for (; k < kiter; ++k) {
    v2f xa, ya, b;
    if (TRANSA) {
      xa.x = ap0[0]; xa.y = ap0[Ms]; ap0 += 4 * Ms;
      ya.x = ap1[0]; ya.y = ap1[Ms]; ap1 += 4 * Ms;
    } else {
      xa = *(const v2f*)(ap0); ap0 += 4;
      ya = *(const v2f*)(ap1); ap1 += 4;
    }
    b.x = bp[0]; b.y = bp[Ns]; bp += 4 * Ns;
    acc0 = wmma4(xa, b, acc0);
    acc1 = wmma4(ya, b, acc1);
  }

  const int cn = nt + r15;
  const int mb = mt + ((lane >> 4) << 3);
  if (cn < N) {
#pragma unroll
    for (int r = 0; r < 8; ++r) {
      const int cm = mb + r;
      if (cm < M) {
        float v = acc0[r];
        if (bias_mode == 1) v += bias[cn];
        else if (bias_mode == 2) v += bias[cm];
        if (do_relu) v = fmaxf(v, 0.f);
        if (Addb) v += Addb[(size_t)cm * N + cn];
        Cb[(size_t)cm * N + cn] = v;
      }
    }
#pragma unroll
    for (int r = 0; r < 8; ++r) {
      const int cm = mb + 16 + r;
      if (cm < M) {
        float v = acc1[r];
        if (bias_mode == 1) v += bias[cn];
        else if (bias_mode == 2) v += bias[cm];
        if (do_relu) v = fmaxf(v, 0.f);
        if (Addb) v += Addb[(size_t)cm * N + cn];
        Cb[(size_t)cm * N + cn] = v;
      }
    }
  }
}

// Encoder GEMM: A row m = concat(P1[m,0:F], P2[m,0:F]) (K = 2F), bias per-col.
// K % 4 == 0 required (K = 36). Indices clamped as above. 16x16 tile per wave.
__global__ __launch_bounds__(32) void k_wmma_gemm_concat2(
    const float* __restrict__ P1, const float* __restrict__ P2, int F_,
    const float* __restrict__ Bw, const float* __restrict__ bias,
    float* __restrict__ C, int M, int N, int K) {
  const int lane = threadIdx.x & 31;
  const int mt = blockIdx.x << 4;
  const int nt = blockIdx.y << 4;
  const int r15 = lane & 15;
  const int khalf = (lane >> 4) << 1;
  int am = mt + r15; if (am > M - 1) am = M - 1;
  int bn = nt + r15; if (bn > N - 1) bn = N - 1;

  const size_t Ns = (size_t)N;
  const float* p1 = P1 + (size_t)am * F_;
  const float* p2 = P2 + (size_t)am * F_;
  const float* bp = Bw + (size_t)khalf * Ns + bn;

  v8f acc = {0.f, 0.f, 0.f, 0.f, 0.f, 0.f, 0.f, 0.f};
  for (int k0 = 0; k0 < K; k0 += 4) {
    const int ka = k0 + khalf;
    const int kb = ka + 1;
    v2f a, b;
    a.x = (ka < F_) ? p1[ka] : p2[ka - F_];
    a.y = (kb < F_) ? p1[kb] : p2[kb - F_];
    b.x = bp[0]; b.y = bp[Ns]; bp += 4 * Ns;
    acc = wmma4(a, b, acc);
  }
  const int cn = nt + r15;
  const int mb = mt + ((lane >> 4) << 3);
  if (cn < N) {
#pragma unroll
    for (int r = 0; r < 8; ++r) {
      const int cm = mb + r;
      if (cm < M) C[(size_t)cm * N + cn] = acc[r] + bias[cn];
    }
  }
}

// ---------------------------------------------------------------------------
// LayerNorm: one wave per row, wave32 shuffle reduction.
// ---------------------------------------------------------------------------
__global__ __launch_bounds__(128) void k_layernorm(
    const float* __restrict__ x, float* __restrict__ y,
    const float* __restrict__ g, const float* __restrict__ b,
    long long rows, int D) {
  const int wid = threadIdx.x >> 5;
  const int lane = threadIdx.x & 31;
  const long long row = (long long)blockIdx.x * 4 + wid;
  if (row >= rows) return;
  const float* xr = x + row * D;
  float s = 0.f, ss = 0.f;
  for (int c = lane; c < D; c += 32) { float v = xr[c]; s += v; ss += v * v; }
#pragma unroll
  for (int off = 16; off > 0; off >>= 1) {
    s += __shfl_xor(s, off, 32);
    ss += __shfl_xor(ss, off, 32);
  }
  const float m = s / (float)D;
  const float var = ss / (float)D - m * m;
  const float inv = rsqrtf(var + 1e-5f);
  float* yr = y + row * D;
  for (int c = lane; c < D; c += 32) yr[c] = (xr[c] - m) * inv * g[c] + b[c];
}

// ---------------------------------------------------------------------------
// Elementwise / scatter helpers (grid-stride).
// ---------------------------------------------------------------------------
#define GS_LOOP(i, total)                                                     \
  for (long long i = (long long)blockIdx.x * blockDim.x + threadIdx.x;        \
       i < (total); i += (long long)gridDim.x * blockDim.x)

__global__ void k_fill0(float* p, long long n) { GS_LOOP(i, n) p[i] = 0.f; }

__global__ void k_count_ids(const int* __restrict__ ids, float* cnt, int n) {
  GS_LOOP(i, n) atomicAdd(&cnt[ids[i]], 1.f);
}
__global__ void k_invert_counts(float* cnt, int n) {
  GS_LOOP(i, n) cnt[i] = 1.f / fmaxf(cnt[i], 1.f);
}

__global__ void k_edge_embed(const float* __restrict__ ew, const int* __restrict__ emap,
                             const float* __restrict__ W, const float* __restrict__ b,
                             float* __restrict__ e, long long total, int C) {
  GS_LOOP(i, total) {
    const int c = (int)(i % C);
    const long long j = i / C;
    e[i] = ew[emap[j]] * W[c] + b[c];
  }
}

__global__ void k_gather_rows(const float* __restrict__ src, float* __restrict__ dst,
                              const int* __restrict__ map, long long total,
                              int Ro, int Ri, int C) {
  GS_LOOP(i, total) {
    const int c = (int)(i % C);
    const long long t = i / C;
    const int r = (int)(t % Ro);
    const long long z = t / Ro;
    dst[i] = src[(z * Ri + map[r]) * C + c];
  }
}

__global__ void k_seg_sum(const float* __restrict__ src, float* __restrict__ dst,
                          const int* __restrict__ ids, long long total,
                          int Ri, int S, int C) {
  GS_LOOP(i, total) {
    const int c = (int)(i % C);
    const long long t = i / C;
    const int r = (int)(t % Ri);
    const long long z = t / Ri;
    atomicAdd(&dst[(z * S + ids[r]) * C + c], src[i]);
  }
}

__global__ void k_scale_rows(float* __restrict__ p, const float* __restrict__ inv,
                             long long total, int S, int C) {
  GS_LOOP(i, total) p[i] *= inv[(int)((i / C) % S)];
}

__global__ void k_edge_msg_agg(const float* __restrict__ x, const float* __restrict__ e,
                               const int* __restrict__ srci, const int* __restrict__ dsti,
                               float* __restrict__ agg, long long total,
                               int E, int Ns, int C) {
  GS_LOOP(i, total) {
    const int c = (int)(i % C);
    const long long t = i / C;
    const int j = (int)(t % E);
    const long long z = t / E;
    const float v = fmaxf(x[(z * Ns + srci[j]) * C + c] + e[(long long)j * C + c], 0.f);
    atomicAdd(&agg[(z * Ns + dsti[j]) * C + c], v);
  }
}

__global__ void k_add_inplace(float* __restrict__ x, const float* __restrict__ y, long long n) {
  GS_LOOP(i, n) x[i] += y[i];
}

__global__ void k_bn_relu_add(float* __restrict__ x, const float* __restrict__ h,
                              const float* __restrict__ g, const float* __restrict__ b,
                              long long total, int C) {
  GS_LOOP(i, total) {
    const int c = (int)(i % C);
    x[i] += fmaxf(h[i] * g[c] + b[c], 0.f);
  }
}

__global__ void k_feat_gather(const float* __restrict__ px, const float* __restrict__ nx,
                              const int* __restrict__ n2p, float* __restrict__ feat,
                              long long total, int T, int N, int P_, int Dp, int Dn) {
  const int C = Dp + Dn;
  GS_LOOP(i, total) {
    const int c = (int)(i % C);
    const long long t1 = i / C;
    const int t = (int)(t1 % T);
    const long long t2 = t1 / T;
    const int n = (int)(t2 % N);
    const int b = (int)(t2 / N);
    float v;
    if (c < Dp) v = px[(((long long)b * T + t) * P_ + n2p[n]) * Dp + c];
    else        v = nx[(((long long)b * T + t) * N + n) * Dn + (c - Dp)];
    feat[i] = v;
  }
}

// ---------------------------------------------------------------------------
// Host-side launch helpers
// ---------------------------------------------------------------------------
static inline dim3 ew_grid(long long total) {
  long long b = (total + 255) / 256;
  if (b > 262144) b = 262144;
  if (b < 1) b = 1;
  return dim3((unsigned)b);
}

static inline void launch_gemm(hipStream_t st, const float* A, const float* Bw,
                               const float* bias, const float* add, float* C,
                               int M, int N, int K, long long sA, long long sB,
                               long long sC, int transA, int bias_mode, int relu,
                               int batches) {
  dim3 g((unsigned)((M + 31) / 32), (unsigned)((N + 15) / 16), (unsigned)batches);
  if (transA)
    k_wmma_gemm_f32<1><<<g, 32, 0, st>>>(A, Bw, bias, add, C, M, N, K, sA, sB,
                                         sC, bias_mode, relu);
  else
    k_wmma_gemm_f32<0><<<g, 32, 0, st>>>(A, Bw, bias, add, C, M, N, K, sA, sB,
                                         sC, bias_mode, relu);
}

static inline void launch_ln(hipStream_t st, const float* x, float* y,
                             const float* g, const float* b, long long rows, int D) {
  k_layernorm<<<(unsigned)((rows + 3) / 4), 128, 0, st>>>(x, y, g, b, rows, D);
}

// MLP-mixer forward over x (NB batches of S x D), params at leaf index pi.
static void run_mixer(hipStream_t st, void* const* d_in, int pi, float* x,
                      float* lnb, float* zb, float* hb, int NB, int S, int D,
                      int HT, int HC, int nlayer) {
  const long long rows = (long long)NB * S;
  for (int l = 0; l < nlayer; ++l) {
    const float* g1  = (const float*)d_in[pi + 0];
    const float* b1  = (const float*)d_in[pi + 1];
    const float* t1W = (const float*)d_in[pi + 2];  // (S, HT)
    const float* t1b = (const float*)d_in[pi + 3];  // (HT,)
    const float* t2W = (const float*)d_in[pi + 4];  // (HT, S)
    const float* t2b = (const float*)d_in[pi + 5];  // (S,)
    const float* g2  = (const float*)d_in[pi + 6];
    const float* b2  = (const float*)d_in[pi + 7];
    const float* c1W = (const float*)d_in[pi + 8];  // (D, HC)
    const float* c1b = (const float*)d_in[pi + 9];
    const float* c2W = (const float*)d_in[pi + 10]; // (HC, D)
    const float* c2b = (const float*)d_in[pi + 11];
    pi += 12;
    // y = LN(x)
    launch_ln(st, x, lnb, g1, b1, rows, D);
    // z = relu(t1W^T @ y + t1b[:,None])   (batched over NB)
    launch_gemm(st, t1W, lnb, t1b, nullptr, zb, HT, D, S,
                0, (long long)S * D, (long long)HT * D, 1, 2, 1, NB);
    // x = x + (t2W^T @ z + t2b[:,None])
    launch_gemm(st, t2W, zb, t2b, x, x, S, D, HT,
                0, (long long)HT * D, (long long)S * D, 1, 2, 0, NB);
    // y2 = LN(x)
    launch_ln(st, x, lnb, g2, b2, rows, D);
    // h = relu(y2 @ c1W + c1b)
    launch_gemm(st, lnb, c1W, c1b, nullptr, hb, (int)rows, HC, D,
                0, 0, 0, 0, 1, 1, 1);
    // x = x + (h @ c2W + c2b)
    launch_gemm(st, hb, c2W, c2b, x, x, (int)rows, D, HC,
                0, 0, 0, 0, 1, 0, 1);
  }
  launch_ln(st, x, x, (const float*)d_in[pi], (const float*)d_in[pi + 1], rows, D);
}

// ---------------------------------------------------------------------------
// Entry point
// ---------------------------------------------------------------------------
extern "C" void kernel_launch(void* const* d_in, const int* in_sizes, int n_in,
                              void* d_out, int out_size, void* d_ws, size_t ws_size,
                              hipStream_t stream) {
  (void)in_sizes; (void)n_in; (void)out_size;
  // Problem dims (fixed by reference setup_inputs)
  const int B = 4, T = 12, N = 2000, F = 18;
  const int Dp = 128, Dn = 64, P_ = 32, Nsub = 4000, Esub = 16000;
  const int NB = B * T;  // 48 (b,t) batches

  // Tensor inputs
  const float* x_raw   = (const float*)d_in[0];
  const float* valid_x = (const float*)d_in[1];
  const float* edge_w  = (const float*)d_in[2];
  const int* nodes_map = (const int*)d_in[3];
  const int* edges_map = (const int*)d_in[4];
  const int* e_src     = (const int*)d_in[5];          // edge_index[0]
  const int* e_dst     = (const int*)d_in[5] + Esub;   // edge_index[1]
  const int* sgb       = (const int*)d_in[6];
  const int* n2p       = (const int*)d_in[7];
  // Param leaves in setup_inputs() dict-insertion order
  enum {
    P_ENCP_W = 8, P_ENCP_B, P_ENCN_W, P_ENCN_B, P_EDGE_W, P_EDGE_B,
    P_G0_LW, P_G0_LB, P_G0_BG, P_G0_BB,
    P_G1_LW, P_G1_LB, P_G1_BG, P_G1_BB,
    P_U0_W, P_U0_B,
    P_MIXP = 24,   // 26 leaves
    P_MIXN = 50,   // 26 leaves
    P_R_L1W = 76, P_R_L1B, P_R_L2W, P_R_L2B
  };

  // Workspace carve-up (floats)
  float* W = (float*)d_ws;
  size_t o = 0;
  float* x_enc  = W + o; o += (size_t)NB * N * Dp;        // 12.288M  (reused: nsum)
  float* nodesx = W + o; o += (size_t)NB * N * Dn;        //  6.144M
  float* x_sub  = W + o; o += (size_t)NB * Nsub * Dp;     // 24.576M
  float* e_sub  = W + o; o += (size_t)Esub * Dp;          //  2.048M  (reused: rbuf)
  float* agg    = W + o; o += (size_t)NB * Nsub * Dp;     // 24.576M  (reused: sub_g, feat)
  float* hbuf   = W + o; o += (size_t)NB * Nsub * Dp;     // 24.576M
  float* patchx = W + o; o += (size_t)NB * P_ * Dp;
  float* lnp    = W + o; o += (size_t)NB * P_ * Dp;
  float* zp     = W + o; o += (size_t)NB * (2 * P_) * Dp;
  float* lnn    = W + o; o += (size_t)NB * N * Dn;
  float* zn     = W + o; o += (size_t)NB * (N / 2) * Dn;
  float* cdst   = W + o; o += Nsub;
  float* csub   = W + o; o += P_;
  float* cnode  = W + o; o += N;
  if (ws_size < o * sizeof(float)) return;  // workspace too small: bail safely

  // ---- segment counts (clip>=1, inverted) -------------------------------
  k_fill0<<<ew_grid(Nsub + P_ + N), 256, 0, stream>>>(cdst, Nsub + P_ + N);
  k_count_ids<<<ew_grid(Esub), 256, 0, stream>>>(e_dst, cdst, Esub);
  k_count_ids<<<ew_grid(Nsub), 256, 0, stream>>>(sgb, csub, Nsub);
  k_count_ids<<<ew_grid(Nsub), 256, 0, stream>>>(nodes_map, cnode, Nsub);
  k_invert_counts<<<ew_grid(Nsub + P_ + N), 256, 0, stream>>>(cdst, Nsub + P_ + N);

  // ---- encoders: concat(x_raw, valid_x) @ W + b -------------------------
  {
    const int M = NB * N, K = 2 * F;
    dim3 gp((M + 15) / 16, (Dp + 15) / 16);
    k_wmma_gemm_concat2<<<gp, 32, 0, stream>>>(x_raw, valid_x, F,
        (const float*)d_in[P_ENCP_W], (const float*)d_in[P_ENCP_B], x_enc, M, Dp, K);
    dim3 gn((M + 15) / 16, (Dn + 15) / 16);
    k_wmma_gemm_concat2<<<gn, 32, 0, stream>>>(x_raw, valid_x, F,
        (const float*)d_in[P_ENCN_W], (const float*)d_in[P_ENCN_B], nodesx, M, Dn, K);
  }

  // ---- gather to subgraph nodes; edge embedding -------------------------
  const long long subElems = (long long)NB * Nsub * Dp;
  k_gather_rows<<<ew_grid(subElems), 256, 0, stream>>>(x_enc, x_sub, nodes_map,
                                                       subElems, Nsub, N, Dp);
  k_edge_embed<<<ew_grid((long long)Esub * Dp), 256, 0, stream>>>(
      edge_w, edges_map, (const float*)d_in[P_EDGE_W], (const float*)d_in[P_EDGE_B],
      e_sub, (long long)Esub * Dp, Dp);

  // ---- GNN layers -------------------------------------------------------
  const long long msgElems = (long long)NB * Esub * Dp;
  for (int gi = 0; gi < 2; ++gi) {
    if (gi > 0) {
      // sub = seg_mean(x, sgb, P)[sgb]; x += relu(sub @ U + b)
      k_fill0<<<ew_grid((long long)NB * P_ * Dp), 256, 0, stream>>>(patchx, (long long)NB * P_ * Dp);
      k_seg_sum<<<ew_grid(subElems), 256, 0, stream>>>(x_sub, patchx, sgb, subElems, Nsub, P_, Dp);
      k_scale_rows<<<ew_grid((long long)NB * P_ * Dp), 256, 0, stream>>>(patchx, csub,
                                                                         (long long)NB * P_ * Dp, P_, Dp);
      k_gather_rows<<<ew_grid(subElems), 256, 0, stream>>>(patchx, agg, sgb, subElems, Nsub, P_, Dp);
      launch_gemm(stream, agg, (const float*)d_in[P_U0_W], (const float*)d_in[P_U0_B],
                  x_sub, x_sub, NB * Nsub, Dp, Dp, 0, 0, 0, 0, 1, 1, 1);
      // x = seg_mean(x, nodes_map, N)[nodes_map]  (nsum aliases x_enc)
      k_fill0<<<ew_grid((long long)NB * N * Dp), 256, 0, stream>>>(x_enc, (long long)NB * N * Dp);
      k_seg_sum<<<ew_grid(subElems), 256, 0, stream>>>(x_sub, x_enc, nodes_map, subElems, Nsub, N, Dp);
      k_scale_rows<<<ew_grid((long long)NB * N * Dp), 256, 0, stream>>>(x_enc, cnode,
                                                                        (long long)NB * N * Dp, N, Dp);
      k_gather_rows<<<ew_grid(subElems), 256, 0, stream>>>(x_enc, x_sub, nodes_map, subElems, Nsub, N, Dp);
    }
    // agg = seg_mean(relu(x[src] + e), dst, Nsub)
    k_fill0<<<ew_grid(subElems), 256, 0, stream>>>(agg, subElems);
    k_edge_msg_agg<<<ew_grid(msgElems), 256, 0, stream>>>(x_sub, e_sub, e_src, e_dst,
                                                          agg, msgElems, Esub, Nsub, Dp);
    k_scale_rows<<<ew_grid(subElems), 256, 0, stream>>>(agg, cdst, subElems, Nsub, Dp);
    // h = (x + agg) @ lin.W + b ; x += relu(h*bn_g + bn_b)
    k_add_inplace<<<ew_grid(subElems), 256, 0, stream>>>(agg, x_sub, subElems);
    const int LW = (gi == 0) ? P_G0_LW : P_G1_LW;
    launch_gemm(stream, agg, (const float*)d_in[LW], (const float*)d_in[LW + 1],
                nullptr, hbuf, NB * Nsub, Dp, Dp, 0, 0, 0, 0, 1, 0, 1);
    k_bn_relu_add<<<ew_grid(subElems), 256, 0, stream>>>(x_sub, hbuf,
        (const float*)d_in[LW + 2], (const float*)d_in[LW + 3], subElems, Dp);
  }

  // ---- patch_x = seg_mean(x, sgb, P) ------------------------------------
  k_fill0<<<ew_grid((long long)NB * P_ * Dp), 256, 0, stream>>>(patchx, (long long)NB * P_ * Dp);
  k_seg_sum<<<ew_grid(subElems), 256, 0, stream>>>(x_sub, patchx, sgb, subElems, Nsub, P_, Dp);
  k_scale_rows<<<ew_grid((long long)NB * P_ * Dp), 256, 0, stream>>>(patchx, csub,
                                                                     (long long)NB * P_ * Dp, P_, Dp);

  // ---- mixers -----------------------------------------------------------
  run_mixer(stream, d_in, P_MIXP, patchx, lnp, zp, hbuf, NB, P_, Dp, 2 * P_, 4 * Dp, 2);
  run_mixer(stream, d_in, P_MIXN, nodesx, lnn, zn, hbuf, NB, N, Dn, N / 2, 4 * Dn, 2);

  // ---- readout ----------------------------------------------------------
  const int Cf = Dp + Dn;                       // 192
  const long long featElems = (long long)B * N * T * Cf;
  float* feat = agg;                            // reuse (18.4M <= 24.6M)
  float* rbuf = e_sub;                          // reuse (8000*256 == 2.048M)
  k_feat_gather<<<ew_grid(featElems), 256, 0, stream>>>(patchx, nodesx, n2p, feat,
                                                        featElems, T, N, P_, Dp, Dn);
  launch_gemm(stream, feat, (const float*)d_in[P_R_L1W], (const float*)d_in[P_R_L1B],
              nullptr, rbuf, B * N, 256, T * Cf, 0, 0, 0, 0, 1, 1, 1);
  launch_gemm(stream, rbuf, (const float*)d_in[P_R_L2W], (const float*)d_in[P_R_L2B],
              nullptr, (float*)d_out, B * N, 12, 256, 0, 0, 0, 0, 1, 0, 1);
}